// ScatteredExperts_9414568313163
// MI455X (gfx1250) — compile-verified
//
#include <hip/hip_runtime.h>

typedef float v2f __attribute__((ext_vector_type(2)));
typedef float v8f __attribute__((ext_vector_type(8)));

// Problem constants (match reference)
#define N_TOK 4096
#define TOPK  2
#define NEXP  8
#define INF   2048
#define OUTF  2048
#define LROWS (N_TOK * TOPK)   // 8192

// Tiling
#define KC    32               // k-chunk staged in LDS (double buffered)
#define BN    128              // output cols per workgroup
#define WN    32               // output cols per wave (2x 16-wide WMMA tiles)
#define APAD  4
#define BPAD  4

// Generic (__shared__) pointer -> 32-bit LDS byte address for async-to-LDS ops
#define LDS_ADDR(p) ((unsigned)(uintptr_t)(__attribute__((address_space(3))) const void*)(p))

__device__ __forceinline__ void async_copy_b128(unsigned lds_addr, const float* gptr) {
    unsigned long long ga = (unsigned long long)(uintptr_t)gptr;
    // GV mode: vdst = LDS byte address, vaddr = 64-bit global address, no SADDR.
    asm volatile("global_load_async_to_lds_b128 %0, %1, off"
                 :: "v"(lds_addr), "v"(ga) : "memory");
}

__device__ __forceinline__ void wait_async_all() {
    asm volatile("s_wait_asynccnt 0x0" ::: "memory");
}

__global__ __launch_bounds__(128)
void moe_wmma_f32_kernel(const float* __restrict__ x,        // [N_TOK, INF]
                         const float* __restrict__ w,        // [NEXP, INF, OUTF]
                         const int*   __restrict__ bin_ids,  // [LROWS] sorted expert ids
                         const int*   __restrict__ indices,  // [LROWS] flat routed index
                         const float* __restrict__ gates,    // [N_TOK, TOPK]
                         float*       __restrict__ out)      // [N_TOK, OUTF]
{
    __shared__ float As[2][16][KC + APAD];    // gathered activations (double buffered)
    __shared__ float Bs[2][KC][BN + BPAD];    // expert weight tile (double buffered)
    __shared__ int   tok_s[16];
    __shared__ int   exp_s[16];
    __shared__ float gate_s[16];

    const int nTiles = OUTF / BN;             // 16
    const int mBase  = (blockIdx.x / nTiles) * 16;
    const int nBase  = (blockIdx.x % nTiles) * BN;

    const int tid  = threadIdx.x;
    const int wave = tid >> 5;
    const int lane = tid & 31;
    const int half = lane >> 4;               // 0: lanes 0-15, 1: lanes 16-31
    const int lm   = lane & 15;

    if (tid < 16) {
        int idx = indices[mBase + tid];
        int tk  = idx / TOPK;
        int sl  = idx - tk * TOPK;
        tok_s[tid]  = tk;
        gate_s[tid] = gates[tk * TOPK + sl];
        exp_s[tid]  = bin_ids[mBase + tid];
    }
    __syncthreads();

    const int eLo = exp_s[0];
    const int eHi = exp_s[15];                // sorted => tile spans [eLo, eHi]

    // Per-thread staging geometry (all b128, 16B aligned: pads are multiples of 4 floats)
    const int ar  = tid >> 3;                 // A row 0..15   (KC/4 = 8 slots/row)
    const int ac4 = (tid & 7) << 2;           // A col 0..28

    for (int e = eLo; e <= eHi; ++e) {
        const float* __restrict__ we = w + (size_t)e * INF * OUTF;
        v8f acc0 = {};
        v8f acc1 = {};

        __syncthreads();                      // prior expert's LDS reads complete

        // ---- prologue: async-stage chunk 0 into buffer 0 ----
        async_copy_b128(LDS_ADDR(&As[0][ar][ac4]),
                        x + (size_t)tok_s[ar] * INF + ac4);
        #pragma unroll
        for (int i = 0; i < 8; ++i) {
            int s  = tid + i * 128;           // 1024 float4 slots of B
            int r  = s >> 5;                  // row 0..31
            int c4 = (s & 31) << 2;           // col 0..124 step 4
            async_copy_b128(LDS_ADDR(&Bs[0][r][c4]),
                            we + (size_t)r * OUTF + nBase + c4);
        }

        for (int kc = 0; kc < INF; kc += KC) {
            const int buf = (kc / KC) & 1;

            wait_async_all();                 // my async stores into LDS landed
            __syncthreads();                  // everyone's landed; prev buf free

            // ---- async-stage next chunk into the other buffer ----
            if (kc + KC < INF) {
                const int nb = buf ^ 1;
                const int kn = kc + KC;
                async_copy_b128(LDS_ADDR(&As[nb][ar][ac4]),
                                x + (size_t)tok_s[ar] * INF + kn + ac4);
                #pragma unroll
                for (int i = 0; i < 8; ++i) {
                    int s  = tid + i * 128;
                    int r  = s >> 5;
                    int c4 = (s & 31) << 2;
                    async_copy_b128(LDS_ADDR(&Bs[nb][r][c4]),
                                    we + (size_t)(kn + r) * OUTF + nBase + c4);
                }
            }

            // ---- 8 WMMA k-steps of 4, two 16-wide N sub-tiles per wave ----
            #pragma unroll
            for (int ks = 0; ks < KC; ks += 4) {
                const int ka = ks + half * 2; // lanes 16-31 carry K+2,K+3
                v2f a, b0, b1;
                a.x  = As[buf][lm][ka];
                a.y  = As[buf][lm][ka + 1];
                b0.x = Bs[buf][ka][wave * WN + lm];
                b0.y = Bs[buf][ka + 1][wave * WN + lm];
                b1.x = Bs[buf][ka][wave * WN + 16 + lm];
                b1.y = Bs[buf][ka + 1][wave * WN + 16 + lm];
                acc0 = __builtin_amdgcn_wmma_f32_16x16x4_f32(
                           false, a, false, b0, (short)0, acc0, false, false);
                acc1 = __builtin_amdgcn_wmma_f32_16x16x4_f32(
                           false, a, false, b1, (short)0, acc1, false, false);
            }
        }

        // ---- gated scatter-add; D layout: n=lm, m = vgpr + 8*half ----
        #pragma unroll
        for (int r = 0; r < 8; ++r) {
            const int m = r + half * 8;
            if (exp_s[m] == e) {
                const float g = gate_s[m];
                float* op = out + (size_t)tok_s[m] * OUTF + nBase + wave * WN;
                atomicAdd(op + lm,      g * acc0[r]);
                atomicAdd(op + 16 + lm, g * acc1[r]);
            }
        }
    }
}

__global__ __launch_bounds__(256)
void zero_out_kernel(float4* __restrict__ out, int n4)
{
    int i = blockIdx.x * blockDim.x + threadIdx.x;
    if (i < n4) out[i] = make_float4(0.f, 0.f, 0.f, 0.f);
}

extern "C" void kernel_launch(void* const* d_in, const int* in_sizes, int n_in,
                              void* d_out, int out_size, void* d_ws, size_t ws_size,
                              hipStream_t stream) {
    const float* x        = (const float*)d_in[0];
    const float* w        = (const float*)d_in[1];
    const int*   bin_ids  = (const int*)d_in[2];
    const int*   indices  = (const int*)d_in[3];
    // d_in[4] padded_block_idxs: schedule metadata, math-inert -> unused
    // d_in[5] expert_offsets: redundant with per-row bin_ids -> unused
    const float* gates    = (const float*)d_in[6];
    float*       out      = (float*)d_out;

    const int n4 = (N_TOK * OUTF) / 4;
    zero_out_kernel<<<(n4 + 255) / 256, 256, 0, stream>>>((float4*)out, n4);

    const int grid = (LROWS / 16) * (OUTF / BN);   // 512 * 16 = 8192 blocks
    moe_wmma_f32_kernel<<<grid, 128, 0, stream>>>(x, w, bin_ids, indices, gates, out);
}